// SwitchModel_83408264888609
// MI455X (gfx1250) — compile-verified
//
#include <hip/hip_runtime.h>

typedef __attribute__((ext_vector_type(2))) float v2f;
typedef __attribute__((ext_vector_type(8))) float v8f;

// ---------------------------------------------------------------------------
// Zero a float buffer.
__global__ void sage_zero_f32(float* __restrict__ p, int n) {
  int i = blockIdx.x * blockDim.x + threadIdx.x;
  if (i < n) p[i] = 0.0f;
}

// ---------------------------------------------------------------------------
// Out-degree of src[] into deg1, out-degree of dst[] (reverse graph) into deg2.
__global__ void sage_degree(const int* __restrict__ src, const int* __restrict__ dst,
                            float* __restrict__ deg1, float* __restrict__ deg2, int E) {
  int e = blockIdx.x * blockDim.x + threadIdx.x;
  if (e < E) {
    atomicAdd(&deg1[src[e]], 1.0f);
    atomicAdd(&deg2[dst[e]], 1.0f);
  }
}

// ---------------------------------------------------------------------------
// In-place: deg1 -> 1/max(deg1,1), deg2 -> 1/max(deg2,1), inv3 = 1/max(deg1+deg2,1).
__global__ void sage_invdeg(float* __restrict__ d1, float* __restrict__ d2,
                            float* __restrict__ inv3, int n) {
  int i = blockIdx.x * blockDim.x + threadIdx.x;
  if (i < n) {
    float a = d1[i], b = d2[i];
    d1[i]  = 1.0f / fmaxf(a, 1.0f);
    d2[i]  = 1.0f / fmaxf(b, 1.0f);
    inv3[i] = 1.0f / fmaxf(a + b, 1.0f);
  }
}

// ---------------------------------------------------------------------------
// Edge scatter: agg[dst] += x[src] * invdeg[src].  16 threads per edge,
// float4 (16B) gathers, 4 f32 atomics per thread (resolve in L2; the 25.6MB
// agg buffer fits the 192MB L2).
__global__ void sage_scatter(const float* __restrict__ x, const int* __restrict__ src,
                             const int* __restrict__ dst, const float* __restrict__ invdeg,
                             float* __restrict__ agg, int E) {
  int gid = blockIdx.x * blockDim.x + threadIdx.x;
  int e = gid >> 4;
  if (e >= E) return;
  int part = gid & 15;
  int s = src[e];
  int d = dst[e];
  float w = invdeg[s];
  float4 v = ((const float4*)(x + (size_t)s * 64))[part];
  float* ar = agg + (size_t)d * 64 + part * 4;
  atomicAdd(ar + 0, v.x * w);
  atomicAdd(ar + 1, v.y * w);
  atomicAdd(ar + 2, v.z * w);
  atomicAdd(ar + 3, v.w * w);
}

// ---------------------------------------------------------------------------
// h[N,64] = act( [x | agg] (N x 128) @ W (128 x 64) + b ), act = ReLU optional.
// 8 waves/block; each wave computes a 16x64 stripe with V_WMMA_F32_16X16X4_F32.
// W + bias staged in LDS once per block.
//
// fp32 A-fragment layout (ISA 7.12.2, 32-bit A 16x4): lanes 0-15 hold K=k,k+1
// of row M=lane; lanes 16-31 hold K=k+2,k+3 of row M=lane-16.
// fp32 B-fragment (4x16, row striped across lanes per VGPR): VGPR0 = rows
// {k, k+2}, VGPR1 = rows {k+1, k+3} across the two lane halves.
// C/D layout: VGPR j -> row j (lanes 0-15) / row j+8 (lanes 16-31), col = lane&15.
__global__ __launch_bounds__(256)
void sage_gemm_wmma(const float* __restrict__ x, const float* __restrict__ agg,
                    const float* __restrict__ W, const float* __restrict__ bias,
                    float* __restrict__ h, int nrows, int do_relu) {
  __shared__ float Wl[128 * 64];
  __shared__ float bl[64];

  int tid = threadIdx.x;
  {
    const float4* W4 = (const float4*)W;
    float4* Wl4 = (float4*)Wl;
    for (int i = tid; i < 2048; i += 256) Wl4[i] = W4[i];
    if (tid < 64) bl[tid] = bias[tid];
  }
  __syncthreads();

  int wave = tid >> 5;
  int lane = tid & 31;
  int rowBase = blockIdx.x * 128 + wave * 16;
  if (rowBase >= nrows) return;  // N is a multiple of 16, whole-wave guard

  int lhalf = lane >> 4;   // 0: lanes 0-15, 1: lanes 16-31
  int l15   = lane & 15;
  int arow  = rowBase + l15;
  const float* xrow = x   + (size_t)arow * 64;
  const float* grow = agg + (size_t)arow * 64;

  v8f acc[4];
  acc[0] = v8f{}; acc[1] = v8f{}; acc[2] = v8f{}; acc[3] = v8f{};

  for (int k = 0; k < 128; k += 4) {
    // A fragment: concat-matrix column pair for this lane half
    int koff = k + (lhalf ? 2 : 0);
    const float* srcr = (koff < 64) ? xrow : grow;
    int cc = koff & 63;
    v2f a;
    a.x = srcr[cc];
    a.y = srcr[cc + 1];

    int kb0 = (k + (lhalf ? 2 : 0)) * 64;  // B VGPR0 row
    int kb1 = (k + (lhalf ? 3 : 1)) * 64;  // B VGPR1 row

#pragma unroll
    for (int nt = 0; nt < 4; ++nt) {
      int col = nt * 16 + l15;
      v2f b;
      b.x = Wl[kb0 + col];
      b.y = Wl[kb1 + col];
      acc[nt] = __builtin_amdgcn_wmma_f32_16x16x4_f32(
          /*neg_a=*/false, a, /*neg_b=*/false, b,
          /*c_mod=*/(short)0, acc[nt], /*reuse_a=*/false, /*reuse_b=*/false);
    }
  }

  // Epilogue: bias + optional ReLU, store per C/D layout.
  float* hrow = h + (size_t)rowBase * 64;
#pragma unroll
  for (int nt = 0; nt < 4; ++nt) {
    int col = nt * 16 + l15;
    float bv = bl[col];
#pragma unroll
    for (int j = 0; j < 8; ++j) {
      int r = j + (lhalf ? 8 : 0);
      float v = acc[nt][j] + bv;
      if (do_relu) v = fmaxf(v, 0.0f);
      hrow[(size_t)r * 64 + col] = v;
    }
  }
}

// ---------------------------------------------------------------------------
extern "C" void kernel_launch(void* const* d_in, const int* in_sizes, int n_in,
                              void* d_out, int out_size, void* d_ws, size_t ws_size,
                              hipStream_t stream) {
  const float* x   = (const float*)d_in[0];
  const int*   src = (const int*)d_in[1];
  const int*   dst = (const int*)d_in[2];
  const float* W1  = (const float*)d_in[3];
  const float* b1  = (const float*)d_in[4];
  const float* W2  = (const float*)d_in[5];
  const float* b2  = (const float*)d_in[6];
  const float* W3  = (const float*)d_in[7];
  const float* b3  = (const float*)d_in[8];

  const int N = in_sizes[0] / 64;
  const int E = in_sizes[1];
  float* out = (float*)d_out;

  // Workspace layout: A[N*64] | B[N*64] | inv1[N] | inv2[N] | inv3[N]
  float* A    = (float*)d_ws;
  float* B    = A    + (size_t)N * 64;
  float* inv1 = B    + (size_t)N * 64;
  float* inv2 = inv1 + N;
  float* inv3 = inv2 + N;

  const int TB = 256;
  const int nfeat = N * 64;
  const int featBlocks = (nfeat + TB - 1) / TB;
  const int scatBlocks = (E * 16 + TB - 1) / TB;
  const int gemmBlocks = (N + 127) / 128;

  // Degrees (counted in-place in inv1/inv2, then inverted).
  sage_zero_f32<<<(2 * N + TB - 1) / TB, TB, 0, stream>>>(inv1, 2 * N);
  sage_degree<<<(E + TB - 1) / TB, TB, 0, stream>>>(src, dst, inv1, inv2, E);
  sage_invdeg<<<(N + TB - 1) / TB, TB, 0, stream>>>(inv1, inv2, inv3, N);

  // Layer 1: forward graph ('O'), ReLU.   x -> B
  sage_zero_f32<<<featBlocks, TB, 0, stream>>>(A, nfeat);
  sage_scatter<<<scatBlocks, TB, 0, stream>>>(x, src, dst, inv1, A, E);
  sage_gemm_wmma<<<gemmBlocks, TB, 0, stream>>>(x, A, W1, b1, B, N, 1);

  // Layer 2: reversed graph ('I'), ReLU.  B -> out (temp)
  sage_zero_f32<<<featBlocks, TB, 0, stream>>>(A, nfeat);
  sage_scatter<<<scatBlocks, TB, 0, stream>>>(B, dst, src, inv2, A, E);
  sage_gemm_wmma<<<gemmBlocks, TB, 0, stream>>>(B, A, W2, b2, out, N, 1);

  // Layer 3: undirected ('U'), no activation.  out -> out (in-place per-row)
  sage_zero_f32<<<featBlocks, TB, 0, stream>>>(A, nfeat);
  sage_scatter<<<scatBlocks, TB, 0, stream>>>(out, src, dst, inv3, A, E);
  sage_scatter<<<scatBlocks, TB, 0, stream>>>(out, dst, src, inv3, A, E);
  sage_gemm_wmma<<<gemmBlocks, TB, 0, stream>>>(out, A, W3, b3, out, N, 0);
}